// YFCAM_Module_69389491634869
// MI455X (gfx1250) — compile-verified
//
#include <hip/hip_runtime.h>
#include <cstdint>

// MI455X / gfx1250, wave32. Heavy math via V_WMMA_F32_16X16X4_F32 (fp32 to
// match fp32 reference numerics); LDS staging via the Tensor Data Mover
// (tensor_load_to_lds + s_wait_tensorcnt) with per-lane float4 fallback.
//
// Pipeline:
//  K1 qk_proj     : QK[b][16][HW] = [Wq;Wk](16x64) @ rgb(64xHW) + [bq;bk]
//  K2 energy_part : energy[b] = q_view(64x8192) @ k_view^T (K split 4 ways)
//  K3 softmax_G   : softmax(rowmax-energy); G = alpha*(attn@Wv)+I; abeta=alpha*attn@bv
//  K4 out_gemm    : out[b] = G(64x64) @ y(64xHW) + abeta  (v intermediate fused away)
//                   each wave owns TWO M-tiles so every LDS B-fragment feeds
//                   two WMMAs (halves LDS bandwidth per matrix op)

typedef float v2f __attribute__((ext_vector_type(2)));
typedef float v8f __attribute__((ext_vector_type(8)));
typedef unsigned int v4u __attribute__((ext_vector_type(4)));
typedef int v8i __attribute__((ext_vector_type(8)));
typedef int v4i __attribute__((ext_vector_type(4)));

#define WMMA_F32(a, b, c) \
  __builtin_amdgcn_wmma_f32_16x16x4_f32(false, (a), false, (b), (short)0, (c), false, false)

constexpr int Bn = 8, C = 64, HW = 256 * 256;
constexpr int NB = 128;   // n-columns per workgroup block (divides HW)
constexpr int LP = 136;   // LDS row stride (words): 128 data + 8 pad (TDM pad codes 6/7)
constexpr int KS = 4;     // K-split factor for the energy GEMM
constexpr int CH = 256;   // K-chunk staged in LDS per energy iteration
constexpr int KP = 260;   // LDS row stride (words): 256 data + 4 pad (TDM pad codes 7/3)

#if defined(__gfx1250__) && __has_builtin(__builtin_amdgcn_tensor_load_to_lds) && \
    __has_builtin(__builtin_amdgcn_s_wait_tensorcnt)
#define USE_TDM 1
#else
#define USE_TDM 0
#endif

#if USE_TDM
__device__ __forceinline__ unsigned lds_off_of(const void* p) {
  // AS3 pointer value == byte offset within the workgroup's LDS allocation
  return (unsigned)(uintptr_t)(__attribute__((address_space(3))) const void*)p;
}

// 2D TDM load: tile_d1 rows of tile_d0 fp32 elements, global row stride
// `stride0` elements, LDS destination padded by pad codes (interval, amount).
__device__ __forceinline__ void tdm_load_2d(unsigned lds_off, const float* gaddr,
                                            unsigned tensor_d0, unsigned tensor_d1,
                                            unsigned long long stride0,
                                            unsigned tile_d0, unsigned tile_d1,
                                            unsigned pad_interval, unsigned pad_amount) {
  unsigned long long ga = (unsigned long long)(uintptr_t)gaddr;
  v4u g0;
  g0[0] = 1u;                                              // count=1 (user mode)
  g0[1] = lds_off;                                         // lds_addr
  g0[2] = (unsigned)(ga & 0xffffffffu);                    // global_addr[31:0]
  g0[3] = (unsigned)((ga >> 32) & 0x01ffffffu) | (2u << 30); // addr[56:32] | type=2
  v8i g1;
  g1[0] = (int)((2u << 16) |                               // data_size = 4B
                (1u << 20) |                               // pad_enable
                (pad_interval << 22) | (pad_amount << 25));
  g1[1] = (int)((tensor_d0 & 0xffffu) << 16);              // abar=0 | td0[15:0]
  g1[2] = (int)((tensor_d0 >> 16) | ((tensor_d1 & 0xffffu) << 16));
  g1[3] = (int)((tensor_d1 >> 16) | (tile_d0 << 16));
  g1[4] = (int)(tile_d1 & 0xffffu);                        // tile_dim2 = 0 -> 2D
  g1[5] = (int)(stride0 & 0xffffffffu);
  g1[6] = (int)((stride0 >> 32) & 0xffffu);                // stride1 = 0 (unused, 2D)
  g1[7] = 0;
  v4i z4 = {0, 0, 0, 0};
  v8i z8 = {0, 0, 0, 0, 0, 0, 0, 0};
  __builtin_amdgcn_tensor_load_to_lds(g0, g1, z4, z4, z8, 0);
}
#endif

// ---------------------------------------------------------------- K1: q/k proj
__global__ __launch_bounds__(128) void qk_proj(const float* __restrict__ rgb,
                                               const float* __restrict__ Wq,
                                               const float* __restrict__ bq,
                                               const float* __restrict__ Wk,
                                               const float* __restrict__ bk,
                                               float* __restrict__ QK) {
  __shared__ float lds[64 * LP];
  const int tid = threadIdx.x, lane = tid & 31, wv = tid >> 5;
  const int b = blockIdx.y, nbase = blockIdx.x * NB;
  const int m = lane & 15, hh = lane >> 4;

  // A fragments: stacked [Wq; Wk] (16 x 64). Lane m holds row M=m; K = 4j+2*hh+v.
  const float* arow = (m < 8) ? (Wq + m * 64) : (Wk + (m - 8) * 64);
  v2f afrag[16];
#pragma unroll
  for (int j = 0; j < 16; ++j) {
    afrag[j][0] = arow[4 * j + 2 * hh];
    afrag[j][1] = arow[4 * j + 2 * hh + 1];
  }
  float bias[8];
#pragma unroll
  for (int r = 0; r < 8; ++r) {
    int row16 = 8 * hh + r;
    bias[r] = (row16 < 8) ? bq[row16] : bk[row16 - 8];
  }

  const float* src = rgb + (size_t)b * C * HW + nbase;
#if USE_TDM
  if (wv == 0) {                        // one DMA stages the whole 64 x NB tile
    tdm_load_2d(lds_off_of(lds), src, HW, 64, HW, NB, 64, /*pad*/ 6, 7);
    __builtin_amdgcn_s_wait_tensorcnt(0);
  }
#else
#pragma unroll
  for (int it = 0; it < 16; ++it) {
    int idx = tid + it * 128;
    int row = idx >> 5;
    int col = (idx & 31) << 2;
    *(float4*)&lds[row * LP + col] = *(const float4*)&src[(size_t)row * HW + col];
  }
#endif
  __syncthreads();

  float* dst = QK + (size_t)b * 16 * HW + nbase;
#pragma unroll
  for (int t = 0; t < 2; ++t) {
    int nt = wv + 4 * t;
    v8f acc;
#pragma unroll
    for (int r = 0; r < 8; ++r) acc[r] = bias[r];
#pragma unroll
    for (int j = 0; j < 16; ++j) {
      int kk = 4 * j + 2 * hh;
      v2f bf;
      bf[0] = lds[kk * LP + nt * 16 + m];
      bf[1] = lds[(kk + 1) * LP + nt * 16 + m];
      acc = WMMA_F32(afrag[j], bf, acc);
    }
#pragma unroll
    for (int r = 0; r < 8; ++r)          // rows 0-7 -> q, 8-15 -> k (same flat layout)
      dst[(size_t)(8 * hh + r) * HW + nt * 16 + m] = acc[r];
  }
}

// ------------------------------------------------------- K2: energy partials
__global__ __launch_bounds__(128) void energy_partial(const float* __restrict__ QK,
                                                      float* __restrict__ epart) {
  __shared__ float lq[16 * KP];
  __shared__ float lk[16 * KP];
  __shared__ float red[4 * 256];
  const int tid = threadIdx.x, lane = tid & 31, wv = tid >> 5;
  const int tile = blockIdx.x, ks = blockIdx.y, b = blockIdx.z;
  const int tm = tile >> 2, tn = tile & 3;
  const int m = lane & 15, hh = lane >> 4;

  const float* qb = QK + (size_t)b * 16 * HW;  // q_view: 64 x 8192 (flat)
  const float* kb = qb + 8 * HW;               // k_view: 64 x 8192

  v8f acc;
#pragma unroll
  for (int r = 0; r < 8; ++r) acc[r] = 0.f;

  const int kslice = 8192 / KS;
  for (int chunk = 0; chunk < kslice / CH; ++chunk) {
    int k0 = ks * kslice + chunk * CH;
#if USE_TDM
    if (wv == 0) {
      tdm_load_2d(lds_off_of(lq), qb + (size_t)(tm * 16) * 8192 + k0,
                  8192, 16, 8192, CH, 16, /*pad*/ 7, 3);
      tdm_load_2d(lds_off_of(lk), kb + (size_t)(tn * 16) * 8192 + k0,
                  8192, 16, 8192, CH, 16, /*pad*/ 7, 3);
      __builtin_amdgcn_s_wait_tensorcnt(0);
    }
#else
#pragma unroll
    for (int it = 0; it < 8; ++it) {
      int idx = tid + it * 128;
      int row = idx >> 6;
      int col = (idx & 63) << 2;
      *(float4*)&lq[row * KP + col] =
          *(const float4*)&qb[(size_t)(tm * 16 + row) * 8192 + k0 + col];
      *(float4*)&lk[row * KP + col] =
          *(const float4*)&kb[(size_t)(tn * 16 + row) * 8192 + k0 + col];
    }
#endif
    __syncthreads();
#pragma unroll
    for (int jj = 0; jj < CH / 16; ++jj) {
      int kk = wv * (CH / 4) + 4 * jj + 2 * hh;
      v2f af, bf;
      af[0] = lq[m * KP + kk];
      af[1] = lq[m * KP + kk + 1];
      bf[0] = lk[m * KP + kk];
      bf[1] = lk[m * KP + kk + 1];
      acc = WMMA_F32(af, bf, acc);
    }
    __syncthreads();
  }
  // reduce the 4 waves' accumulators, scatter to energy partial slice
#pragma unroll
  for (int r = 0; r < 8; ++r) red[wv * 256 + lane * 8 + r] = acc[r];
  __syncthreads();
  for (int t = tid; t < 256; t += 128) {
    float s = red[t] + red[256 + t] + red[512 + t] + red[768 + t];
    int ln = t >> 3, r = t & 7;
    int row = tm * 16 + 8 * (ln >> 4) + r;
    int col = tn * 16 + (ln & 15);
    epart[((size_t)(ks * Bn + b)) * 4096 + row * 64 + col] = s;
  }
}

// ------------------------------------- K3: reduce + softmax + fold Wv/bv/alpha
__global__ __launch_bounds__(256) void softmax_G(const float* __restrict__ epart,
                                                 const float* __restrict__ Wv,
                                                 const float* __restrict__ bv,
                                                 const float* __restrict__ alpha,
                                                 float* __restrict__ G,
                                                 float* __restrict__ abeta) {
  __shared__ float e[4096];
  __shared__ float wvl[4096];
  const int tid = threadIdx.x, b = blockIdx.x;
  for (int i = tid; i < 4096; i += 256) {
    float s = 0.f;
#pragma unroll
    for (int ks = 0; ks < KS; ++ks) s += epart[((size_t)(ks * Bn + b)) * 4096 + i];
    e[i] = s;
    wvl[i] = Wv[i];
  }
  __syncthreads();
  // softmax(rowmax - e) == exp(rowmin - e) / sum(exp(rowmin - e))
  if (tid < 64) {
    int r = tid;
    float mn = e[r * 64];
    for (int d = 1; d < 64; ++d) mn = fminf(mn, e[r * 64 + d]);
    float s = 0.f;
    for (int d = 0; d < 64; ++d) s += expf(mn - e[r * 64 + d]);
    float inv = 1.f / s;
    for (int d = 0; d < 64; ++d) e[r * 64 + d] = expf(mn - e[r * 64 + d]) * inv;
  }
  __syncthreads();
  float al = alpha[0];
  for (int i = tid; i < 4096; i += 256) {     // G = alpha*(attn @ Wv) + I
    int c = i >> 6, ee = i & 63;
    float g = 0.f;
    for (int d = 0; d < 64; ++d) g += e[c * 64 + d] * wvl[d * 64 + ee];
    G[(size_t)b * 4096 + i] = al * g + ((c == ee) ? 1.f : 0.f);
  }
  if (tid < 64) {                             // abeta = alpha*(attn @ bv)
    float s = 0.f;
    for (int d = 0; d < 64; ++d) s += e[tid * 64 + d] * bv[d];
    abeta[b * 64 + tid] = al * s;
  }
}

// ------------------------------------------------- K4: out = G @ y + abeta
// Each wave owns TWO tile_m (32 rows of G in registers) and half the n-tiles:
//   wv&1  -> tile_m pair {0,1} or {2,3}
//   wv>>1 -> nt range 0-3 or 4-7
// so each LDS B-fragment feeds two WMMAs.
__global__ __launch_bounds__(128) void out_gemm(const float* __restrict__ y,
                                                const float* __restrict__ G,
                                                const float* __restrict__ abeta,
                                                float* __restrict__ out) {
  __shared__ float lds[64 * LP];
  const int tid = threadIdx.x, lane = tid & 31, wv = tid >> 5;
  const int b = blockIdx.y, nbase = blockIdx.x * NB;
  const int m = lane & 15, hh = lane >> 4;
  const int tmw = (wv & 1) * 2;        // tile_m pair base
  const int nth = (wv >> 1) * 4;       // n-tile range base

  v2f afrag[2][16];
  float cinit[2][8];
#pragma unroll
  for (int t = 0; t < 2; ++t) {
    const float* grow = G + (size_t)b * 4096 + ((tmw + t) * 16 + m) * 64;
#pragma unroll
    for (int j = 0; j < 16; ++j) {
      afrag[t][j][0] = grow[4 * j + 2 * hh];
      afrag[t][j][1] = grow[4 * j + 2 * hh + 1];
    }
#pragma unroll
    for (int r = 0; r < 8; ++r)
      cinit[t][r] = abeta[b * 64 + (tmw + t) * 16 + 8 * hh + r];
  }

  const float* src = y + (size_t)b * C * HW + nbase;
#if USE_TDM
  if (wv == 0) {
    tdm_load_2d(lds_off_of(lds), src, HW, 64, HW, NB, 64, /*pad*/ 6, 7);
    __builtin_amdgcn_s_wait_tensorcnt(0);
  }
#else
#pragma unroll
  for (int it = 0; it < 16; ++it) {
    int idx = tid + it * 128;
    int row = idx >> 5;
    int col = (idx & 31) << 2;
    *(float4*)&lds[row * LP + col] = *(const float4*)&src[(size_t)row * HW + col];
  }
#endif
  __syncthreads();

  float* dst = out + (size_t)b * C * HW + nbase;
#pragma unroll
  for (int t = 0; t < 4; ++t) {
    int nt = nth + t;
    v8f acc0, acc1;
#pragma unroll
    for (int r = 0; r < 8; ++r) { acc0[r] = cinit[0][r]; acc1[r] = cinit[1][r]; }
#pragma unroll
    for (int j = 0; j < 16; ++j) {
      int kk = 4 * j + 2 * hh;
      v2f bf;
      bf[0] = lds[kk * LP + nt * 16 + m];
      bf[1] = lds[(kk + 1) * LP + nt * 16 + m];
      acc0 = WMMA_F32(afrag[0][j], bf, acc0);   // same B fragment feeds
      acc1 = WMMA_F32(afrag[1][j], bf, acc1);   // both M-tiles
    }
#pragma unroll
    for (int r = 0; r < 8; ++r) {
      dst[(size_t)((tmw + 0) * 16 + 8 * hh + r) * HW + nt * 16 + m] = acc0[r];
      dst[(size_t)((tmw + 1) * 16 + 8 * hh + r) * HW + nt * 16 + m] = acc1[r];
    }
  }
}

// -------------------------------------------------------------------- launch
extern "C" void kernel_launch(void* const* d_in, const int* in_sizes, int n_in,
                              void* d_out, int out_size, void* d_ws, size_t ws_size,
                              hipStream_t stream) {
  (void)in_sizes; (void)n_in; (void)out_size; (void)ws_size;
  const float* rgb   = (const float*)d_in[0];
  const float* y     = (const float*)d_in[1];
  const float* Wq    = (const float*)d_in[2];
  const float* bq    = (const float*)d_in[3];
  const float* Wk    = (const float*)d_in[4];
  const float* bk    = (const float*)d_in[5];
  const float* Wv    = (const float*)d_in[6];
  const float* bv    = (const float*)d_in[7];
  const float* alpha = (const float*)d_in[8];
  float* out = (float*)d_out;

  float* ws    = (float*)d_ws;
  float* QK    = ws;                               // B*16*HW floats (33.5 MB)
  float* epart = QK + (size_t)Bn * 16 * HW;        // KS*B*4096
  float* G     = epart + (size_t)KS * Bn * 4096;   // B*4096
  float* abeta = G + (size_t)Bn * 4096;            // B*64

  qk_proj<<<dim3(HW / NB, Bn), 128, 0, stream>>>(rgb, Wq, bq, Wk, bk, QK);
  energy_partial<<<dim3(16, KS, Bn), 128, 0, stream>>>(QK, epart);
  softmax_G<<<dim3(Bn), 256, 0, stream>>>(epart, Wv, bv, alpha, G, abeta);
  out_gemm<<<dim3(HW / NB, Bn), 128, 0, stream>>>(y, G, abeta, out);
}